// AutoregressiveDecoder_25048249270857
// MI455X (gfx1250) — compile-verified
//
#include <hip/hip_runtime.h>

typedef __attribute__((ext_vector_type(2))) float v2f;
typedef __attribute__((ext_vector_type(8))) float v8f;

#define N_ 512
#define D_ 256
#define H_ 64
#define TS 516   // padded column stride for transposed t in LDS (bank-conflict-free)

static __device__ __forceinline__ v8f wmma_f32_4(v2f a, v2f b, v8f c) {
  // V_WMMA_F32_16X16X4_F32: (neg_a, A, neg_b, B, c_mod, C, reuse_a, reuse_b)
  return __builtin_amdgcn_wmma_f32_16x16x4_f32(false, a, false, b, (short)0, c,
                                               false, false);
}

// ---------------------------------------------------------------------------
// Kernel 1: x = z @ z^T  (512x512x256 SGEMM, f32 WMMA, one 16x16 tile / wave)
// ---------------------------------------------------------------------------
__global__ void __launch_bounds__(128) gram_kernel(const float* __restrict__ z,
                                                   float* __restrict__ out) {
  const int lane = threadIdx.x & 31;
  const int wave = threadIdx.x >> 5;           // 0..3
  const int m    = lane & 15;
  const int kh   = (lane >> 4) << 1;           // K offset 0 or 2
  const int r0   = (blockIdx.y * 4 + wave) * 16;
  const int c0   = blockIdx.x * 16;
  const float* zr = z + (size_t)(r0 + m) * D_;
  const float* zc = z + (size_t)(c0 + m) * D_; // B fragment of z^T == A pattern
  v8f acc = {};
  for (int k0 = 0; k0 < D_; k0 += 4) {
    v2f A = { zr[k0 + kh], zr[k0 + kh + 1] };
    v2f B = { zc[k0 + kh], zc[k0 + kh + 1] };
    acc = wmma_f32_4(A, B, acc);
  }
  const int rbase = r0 + ((lane >> 4) << 3);
#pragma unroll
  for (int p = 0; p < 8; ++p)
    out[(size_t)(rbase + p) * N_ + c0 + m] = acc[p];
}

// ---------------------------------------------------------------------------
// Kernel 2: degs[i,c] = rsqrt(max(colsum(prefix<i), 1)), colsum zeroed c>=i
// ---------------------------------------------------------------------------
__global__ void __launch_bounds__(256) degs_kernel(const float* __restrict__ adj,
                                                   float* __restrict__ degs) {
  const int c = blockIdx.x * 256 + threadIdx.x;   // column, 0..511
  float run = 0.f;
  for (int i = 0; i < N_; ++i) {
    const float cs = (c < i) ? run : 0.f;
    degs[(size_t)i * N_ + c] = __frsqrt_rn(fmaxf(cs, 1.f));
    run += adj[(size_t)i * N_ + c];
  }
}

// ---------------------------------------------------------------------------
// Kernel 3: base = z @ W1[:256]  ([512,256]x[256,64], f32 WMMA)
// ---------------------------------------------------------------------------
__global__ void __launch_bounds__(128) base_kernel(const float* __restrict__ z,
                                                   const float* __restrict__ W1,
                                                   float* __restrict__ base) {
  const int lane = threadIdx.x & 31;
  const int wave = threadIdx.x >> 5;           // h-tile 0..3
  const int m    = lane & 15;
  const int kh   = (lane >> 4) << 1;
  const int r0   = blockIdx.x * 16;
  const int h0   = wave * 16;
  const float* zr = z + (size_t)(r0 + m) * D_;
  v8f acc = {};
  for (int k0 = 0; k0 < D_; k0 += 4) {
    v2f A = { zr[k0 + kh], zr[k0 + kh + 1] };
    v2f B = { W1[(size_t)(k0 + kh) * H_ + h0 + m],
              W1[(size_t)(k0 + kh + 1) * H_ + h0 + m] };
    acc = wmma_f32_4(A, B, acc);
  }
  const int rbase = r0 + ((lane >> 4) << 3);
#pragma unroll
  for (int p = 0; p < 8; ++p)
    base[(size_t)(rbase + p) * H_ + h0 + m] = acc[p];
}

// ---------------------------------------------------------------------------
// Kernel 4: one block per i. Stage t = D_i M_i base (transposed, padded) in
// LDS once; Phase 1 (WMMA, B from LDS): h = relu(D M adj t), v = h @ W2 via
// ds_add_f32. Phase 2: S[i] = D M adj M D v.
// Dynamic LDS: t [H_][TS] + v [N_]  =~ 130 KB.
// ---------------------------------------------------------------------------
__global__ void __launch_bounds__(256) spconv_kernel(const float* __restrict__ adj,
                                                     const float* __restrict__ degs,
                                                     const float* __restrict__ base,
                                                     const float* __restrict__ W2,
                                                     float* __restrict__ S) {
  extern __shared__ float smem[];
  float* t_sh = smem;              // [H_][TS], t_sh[h*TS + c] = deg[c]*base[c,h]
  float* v_sh = smem + H_ * TS;    // [N_]

  const int i    = blockIdx.x;
  const int tid  = threadIdx.x;
  const int lane = tid & 31;
  const int wave = tid >> 5;                   // 0..7
  const int m    = lane & 15;
  const int kh   = (lane >> 4) << 1;

  const float* deg = degs + (size_t)i * N_;

  // Stage transposed scaled base; zero-pad columns [i, cUp) so the K-loop's
  // ragged tail (c up to i+2) reads zeros (prefix mask M folded in here).
  const int cUp = (i + 4 < N_) ? i + 4 : N_;
  for (int idx = tid; idx < cUp * H_; idx += 256) {
    const int c = idx >> 6;
    const int h = idx & (H_ - 1);
    t_sh[h * TS + c] = (c < i) ? deg[c] * base[(size_t)c * H_ + h] : 0.f;
  }
  for (int r = tid; r < N_; r += 256) v_sh[r] = 0.f;
  __syncthreads();

  const float w2_0 = W2[m], w2_1 = W2[16 + m], w2_2 = W2[32 + m], w2_3 = W2[48 + m];

  // Phase 1: row blocks of the i x i region, 16x64 WMMA accumulation each.
  const int nRB = (i + 15) >> 4;
  for (int rb = wave; rb < nRB; rb += 8) {
    const int r0 = rb * 16;
    const float* arow = adj + (size_t)(r0 + m) * N_;
    v8f acc0 = {}, acc1 = {}, acc2 = {}, acc3 = {};
    for (int c0 = 0; c0 < i; c0 += 4) {
      const int ca = c0 + kh;                    // K columns ca, ca+1
      v2f A = { arow[ca], arow[ca + 1] };        // adj garbage ok: t column zeroed
      const float* t0 = t_sh + (size_t)m * TS + ca;            // h-tile 0
      const float* t1 = t_sh + (size_t)(16 + m) * TS + ca;     // h-tile 1
      const float* t2 = t_sh + (size_t)(32 + m) * TS + ca;     // h-tile 2
      const float* t3 = t_sh + (size_t)(48 + m) * TS + ca;     // h-tile 3
      v2f B0 = { t0[0], t0[1] };
      v2f B1 = { t1[0], t1[1] };
      v2f B2 = { t2[0], t2[1] };
      v2f B3 = { t3[0], t3[1] };
      acc0 = wmma_f32_4(A, B0, acc0);
      acc1 = wmma_f32_4(A, B1, acc1);
      acc2 = wmma_f32_4(A, B2, acc2);
      acc3 = wmma_f32_4(A, B3, acc3);
    }
    const int rbase = r0 + ((lane >> 4) << 3);
#pragma unroll
    for (int p = 0; p < 8; ++p) {
      const int r = rbase + p;
      if (r < i) {                               // output mask M (r<i)
        const float dr = deg[r];
        const float contrib = fmaxf(dr * acc0[p], 0.f) * w2_0
                            + fmaxf(dr * acc1[p], 0.f) * w2_1
                            + fmaxf(dr * acc2[p], 0.f) * w2_2
                            + fmaxf(dr * acc3[p], 0.f) * w2_3;
        atomicAdd(&v_sh[r], contrib);            // ds_add_f32
      }
    }
  }
  __syncthreads();

  // dv[c] = deg[c] * v[c], masked to c < i
  for (int c = tid; c < N_; c += 256)
    v_sh[c] = (c < i) ? deg[c] * v_sh[c] : 0.f;
  __syncthreads();

  // Phase 2: S[i,r] = deg[r] * sum_{c<i} adj[r,c] * dv[c]  (adj symmetric ->
  // read adj[c*N+r] for coalesced access across r)
  for (int r = tid; r < N_; r += 256) {
    float sum = 0.f;
    if (r < i) {
      for (int c = 0; c < i; ++c)
        sum += adj[(size_t)c * N_ + r] * v_sh[c];
      sum *= deg[r];
    }
    S[(size_t)i * N_ + r] = sum;
  }
}

// ---------------------------------------------------------------------------
// Kernel 5: out = x + 0.5*(S + S^T)
// ---------------------------------------------------------------------------
__global__ void __launch_bounds__(256) combine_kernel(const float* __restrict__ S,
                                                      float* __restrict__ out) {
  const int idx = blockIdx.x * 256 + threadIdx.x;
  const int i = idx >> 9;
  const int j = idx & (N_ - 1);
  out[idx] += 0.5f * (S[idx] + S[(size_t)j * N_ + i]);
}

extern "C" void kernel_launch(void* const* d_in, const int* in_sizes, int n_in,
                              void* d_out, int out_size, void* d_ws, size_t ws_size,
                              hipStream_t stream) {
  (void)in_sizes; (void)n_in; (void)out_size; (void)ws_size;
  const float* z   = (const float*)d_in[0];   // [512,256]
  const float* adj = (const float*)d_in[1];   // [512,512]
  const float* W1  = (const float*)d_in[2];   // [257,64]
  const float* W2  = (const float*)d_in[3];   // [64,1]
  float* out = (float*)d_out;                 // [512,512]

  char* ws = (char*)d_ws;
  float* degs = (float*)ws;                                              // 1 MB
  float* base = (float*)(ws + (size_t)N_ * N_ * sizeof(float));          // 128 KB
  float* S    = (float*)(ws + (size_t)N_ * N_ * sizeof(float)
                            + (size_t)N_ * H_ * sizeof(float));          // 1 MB

  const size_t spconv_lds = (size_t)(H_ * TS + N_) * sizeof(float);      // ~130 KB
  // Opt in to large dynamic LDS (no-op if not required on this runtime).
  hipFuncSetAttribute(reinterpret_cast<const void*>(spconv_kernel),
                      hipFuncAttributeMaxDynamicSharedMemorySize,
                      (int)spconv_lds);

  gram_kernel  <<<dim3(32, 8), 128, 0, stream>>>(z, out);
  degs_kernel  <<<2,           256, 0, stream>>>(adj, degs);
  base_kernel  <<<32,          128, 0, stream>>>(z, W1, base);
  spconv_kernel<<<N_,          256, (unsigned)spconv_lds, stream>>>(adj, degs, base, W2, S);
  combine_kernel<<<(N_ * N_) / 256, 256, 0, stream>>>(S, out);
}